// FaceClassificationGCN_10058813407638
// MI455X (gfx1250) — compile-verified
//
#include <hip/hip_runtime.h>

typedef float v2f __attribute__((ext_vector_type(2)));
typedef float v8f __attribute__((ext_vector_type(8)));

#define HID 64
#define BN_EPS 1e-5f

// ---------------------------------------------------------------------------
// deg/dinv pipeline
// ---------------------------------------------------------------------------
__global__ void k_deg_init(float* deg, int N) {
    int i = blockIdx.x * blockDim.x + threadIdx.x;
    if (i < N) deg[i] = 1.0f;                      // self loop
}

__global__ void k_deg_count(const int* __restrict__ dst, float* __restrict__ deg, int E) {
    int i = blockIdx.x * blockDim.x + threadIdx.x;
    if (i < E) atomicAdd(&deg[dst[i]], 1.0f);
}

__global__ void k_rsqrt_inplace(float* d, int N) {
    int i = blockIdx.x * blockDim.x + threadIdx.x;
    if (i < N) d[i] = rsqrtf(d[i]);
}

__global__ void k_zero128(float* p) {
    p[threadIdx.x] = 0.0f;                         // <<<1,128>>>
}

// ---------------------------------------------------------------------------
// input layer: h = relu(x @ W_in + b_in), x [N,12], W_in [12,64]
// one thread per (node, feature)
// ---------------------------------------------------------------------------
__global__ void k_input(const float* __restrict__ x, const float* __restrict__ Wi,
                        const float* __restrict__ bi, float* __restrict__ h, int N) {
    int gid = blockIdx.x * blockDim.x + threadIdx.x;
    if (gid >= N * HID) return;
    int n = gid >> 6;
    int f = gid & 63;
    const float* xr = x + (size_t)n * 12;
    float acc = bi[f];
#pragma unroll
    for (int k = 0; k < 12; ++k) acc += xr[k] * Wi[k * HID + f];
    h[gid] = fmaxf(acc, 0.0f);
}

// ---------------------------------------------------------------------------
// hidden GEMM: t = h @ W  (fp32 WMMA 16x16x4), fused self-loop term:
//   agg = t * dinv^2 + b
// One wave computes a 16-row x 64-col strip. 8 waves / block -> 128 rows.
// N must be a multiple of 16 for full waves (100000 = 6250*16); per-lane
// guards still protect stores on a ragged tail.
// ---------------------------------------------------------------------------
__global__ __launch_bounds__(256) void k_gcn_gemm(
    const float* __restrict__ h, const float* __restrict__ W,
    const float* __restrict__ b, const float* __restrict__ dinv,
    float* __restrict__ t, float* __restrict__ agg, int N)
{
    const int lane = threadIdx.x & 31;
    const int wave = threadIdx.x >> 5;
    const int row0 = (blockIdx.x * 8 + wave) * 16;
    if (row0 >= N) return;                         // wave-uniform exit (EXEC stays all-1)

    const int r  = lane & 15;                      // row (A) / col (B,C) within tile
    const int kh = lane >> 4;                      // which K-half this lane holds

    int arow = row0 + r; if (arow > N - 1) arow = N - 1;
    const float* hrow = h + (size_t)arow * HID;

    v8f acc0 = {}, acc1 = {}, acc2 = {}, acc3 = {};

#pragma unroll
    for (int k = 0; k < HID; k += 4) {
        // A frag: lanes 0-15 -> K = k,k+1 ; lanes 16-31 -> K = k+2,k+3
        v2f a = *(const v2f*)(hrow + k + 2 * kh);
        // B frags per 16-col tile, same K striping across lane halves
        const float* w0 = W + (size_t)(k + 2 * kh) * HID + r;
        const float* w1 = w0 + HID;
        v2f b0 = { w0[0],  w1[0]  };
        v2f b1 = { w0[16], w1[16] };
        v2f b2 = { w0[32], w1[32] };
        v2f b3 = { w0[48], w1[48] };
        acc0 = __builtin_amdgcn_wmma_f32_16x16x4_f32(false, a, false, b0, (short)0, acc0, false, false);
        acc1 = __builtin_amdgcn_wmma_f32_16x16x4_f32(false, a, false, b1, (short)0, acc1, false, false);
        acc2 = __builtin_amdgcn_wmma_f32_16x16x4_f32(false, a, false, b2, (short)0, acc2, false, false);
        acc3 = __builtin_amdgcn_wmma_f32_16x16x4_f32(false, a, false, b3, (short)0, acc3, false, false);
    }

    // C/D layout: VGPR v -> row (v + 8*kh), col = n0 + r
#pragma unroll
    for (int v = 0; v < 8; ++v) {
        int rr = row0 + v + 8 * kh;
        if (rr >= N) continue;
        float dv = dinv[rr];
        float sn = dv * dv;                        // self-loop coefficient
        float* tr = t   + (size_t)rr * HID;
        float* ar = agg + (size_t)rr * HID;
        tr[ 0 + r] = acc0[v]; ar[ 0 + r] = acc0[v] * sn + b[ 0 + r];
        tr[16 + r] = acc1[v]; ar[16 + r] = acc1[v] * sn + b[16 + r];
        tr[32 + r] = acc2[v]; ar[32 + r] = acc2[v] * sn + b[32 + r];
        tr[48 + r] = acc3[v]; ar[48 + r] = acc3[v] * sn + b[48 + r];
    }
}

// ---------------------------------------------------------------------------
// edge scatter: agg[dst] += t[src] * dinv[src]*dinv[dst]
// one thread per (edge, 4-feature chunk); float4 gather + 4 f32 atomics.
// Everything is L2-resident (agg = 25.6 MB << 192 MB L2).
// ---------------------------------------------------------------------------
__global__ void k_scatter(const int* __restrict__ src, const int* __restrict__ dst,
                          const float* __restrict__ dinv, const float* __restrict__ t,
                          float* __restrict__ agg, int E)
{
    int gid = blockIdx.x * blockDim.x + threadIdx.x;
    int e = gid >> 4;
    if (e >= E) return;
    int fq = (gid & 15) << 2;
    int s = src[e];
    int d = dst[e];
    float coef = dinv[s] * dinv[d];
    float4 v = *(const float4*)(t + (size_t)s * HID + fq);
    float* p = agg + (size_t)d * HID + fq;
    atomicAdd(p + 0, v.x * coef);
    atomicAdd(p + 1, v.y * coef);
    atomicAdd(p + 2, v.z * coef);
    atomicAdd(p + 3, v.w * coef);
}

// ---------------------------------------------------------------------------
// BatchNorm statistics: per-feature sum & sumsq -> stats[0:64], stats[64:128]
// ---------------------------------------------------------------------------
#define BN_NPB 1024
__global__ __launch_bounds__(256) void k_bnstats(const float* __restrict__ agg,
                                                 float* __restrict__ stats, int N)
{
    __shared__ float s_sum[256];
    __shared__ float s_sq[256];
    int f   = threadIdx.x & 63;
    int sub = threadIdx.x >> 6;                    // 0..3
    int n0  = blockIdx.x * BN_NPB;
    int nend = n0 + BN_NPB; if (nend > N) nend = N;
    float s = 0.0f, q = 0.0f;
    for (int n = n0 + sub; n < nend; n += 4) {
        float v = agg[(size_t)n * HID + f];
        s += v; q += v * v;
    }
    s_sum[threadIdx.x] = s;
    s_sq[threadIdx.x]  = q;
    __syncthreads();
    if (sub == 0) {
        s = s_sum[f] + s_sum[64 + f] + s_sum[128 + f] + s_sum[192 + f];
        q = s_sq[f]  + s_sq[64 + f]  + s_sq[128 + f]  + s_sq[192 + f];
        atomicAdd(&stats[f], s);
        atomicAdd(&stats[64 + f], q);
    }
}

__global__ void k_bnapply(const float* __restrict__ agg, const float* __restrict__ stats,
                          const float* __restrict__ gamma, const float* __restrict__ beta,
                          float* __restrict__ h, int N, int do_relu)
{
    int gid = blockIdx.x * blockDim.x + threadIdx.x;
    if (gid >= N * HID) return;
    int f = gid & 63;
    float invN = 1.0f / (float)N;
    float mu  = stats[f] * invN;
    float var = stats[64 + f] * invN - mu * mu;
    float v = (agg[gid] - mu) * rsqrtf(var + BN_EPS) * gamma[f] + beta[f];
    if (do_relu) v = fmaxf(v, 0.0f);
    h[gid] = v;
}

// ---------------------------------------------------------------------------
// classifier head: out = relu(h @ W_c1 + b_c1) @ W_c2 + b_c2
// one thread per node (64->32->2)
// ---------------------------------------------------------------------------
__global__ __launch_bounds__(256) void k_classify(
    const float* __restrict__ h, const float* __restrict__ Wc1,
    const float* __restrict__ bc1, const float* __restrict__ Wc2,
    const float* __restrict__ bc2, float* __restrict__ out, int N)
{
    int n = blockIdx.x * blockDim.x + threadIdx.x;
    if (n >= N) return;
    const float* hr = h + (size_t)n * HID;
    float c1[32];
#pragma unroll
    for (int j = 0; j < 32; ++j) c1[j] = bc1[j];
    for (int k = 0; k < HID; ++k) {
        float hv = hr[k];
        const float* wr = Wc1 + k * 32;
#pragma unroll
        for (int j = 0; j < 32; ++j) c1[j] += hv * wr[j];
    }
    float o0 = bc2[0], o1 = bc2[1];
#pragma unroll
    for (int j = 0; j < 32; ++j) {
        float a = fmaxf(c1[j], 0.0f);
        o0 += a * Wc2[j * 2 + 0];
        o1 += a * Wc2[j * 2 + 1];
    }
    out[(size_t)n * 2 + 0] = o0;
    out[(size_t)n * 2 + 1] = o1;
}

// ---------------------------------------------------------------------------
// launcher
// ---------------------------------------------------------------------------
extern "C" void kernel_launch(void* const* d_in, const int* in_sizes, int n_in,
                              void* d_out, int out_size, void* d_ws, size_t ws_size,
                              hipStream_t stream)
{
    const float* x     = (const float*)d_in[0];
    const int*   eidx  = (const int*)d_in[1];
    const float* W_in  = (const float*)d_in[2];
    const float* b_in  = (const float*)d_in[3];
    const float* W_g   = (const float*)d_in[4];
    const float* b_g   = (const float*)d_in[5];
    const float* gamma = (const float*)d_in[6];
    const float* beta  = (const float*)d_in[7];
    const float* W_c1  = (const float*)d_in[8];
    const float* b_c1  = (const float*)d_in[9];
    const float* W_c2  = (const float*)d_in[10];
    const float* b_c2  = (const float*)d_in[11];
    float* out = (float*)d_out;

    const int N = in_sizes[0] / 12;
    const int E = in_sizes[1] / 2;
    const int* src = eidx;
    const int* dst = eidx + E;

    // workspace layout (floats)
    float* ws    = (float*)d_ws;
    float* dinv  = ws;                              // N       (deg -> dinv in place)
    float* stats = ws + N;                          // 128
    float* hbuf  = stats + 128;                     // N*64
    float* tbuf  = hbuf + (size_t)N * HID;          // N*64
    float* abuf  = tbuf + (size_t)N * HID;          // N*64

    const int TPB = 256;
    dim3 blk(TPB);

    // degrees -> dinv
    k_deg_init<<<dim3((N + TPB - 1) / TPB), blk, 0, stream>>>(dinv, N);
    k_deg_count<<<dim3((E + TPB - 1) / TPB), blk, 0, stream>>>(dst, dinv, E);
    k_rsqrt_inplace<<<dim3((N + TPB - 1) / TPB), blk, 0, stream>>>(dinv, N);

    // input layer
    k_input<<<dim3((N * HID + TPB - 1) / TPB), blk, 0, stream>>>(x, W_in, b_in, hbuf, N);

    // GCN layers
    const int gemm_blocks = (N + 127) / 128;        // 128 rows per block (8 waves x 16)
    for (int i = 0; i < 3; ++i) {
        k_gcn_gemm<<<dim3(gemm_blocks), blk, 0, stream>>>(
            hbuf, W_g + (size_t)i * HID * HID, b_g + i * HID, dinv, tbuf, abuf, N);
        k_zero128<<<dim3(1), dim3(128), 0, stream>>>(stats);
        {
            long long threads = (long long)E * 16;
            k_scatter<<<dim3((unsigned)((threads + TPB - 1) / TPB)), blk, 0, stream>>>(
                src, dst, dinv, tbuf, abuf, E);
        }
        k_bnstats<<<dim3((N + BN_NPB - 1) / BN_NPB), blk, 0, stream>>>(abuf, stats, N);
        k_bnapply<<<dim3((N * HID + TPB - 1) / TPB), blk, 0, stream>>>(
            abuf, stats, gamma + i * HID, beta + i * HID, hbuf, N, (i < 2) ? 1 : 0);
    }

    // classifier head
    k_classify<<<dim3((N + TPB - 1) / TPB), blk, 0, stream>>>(
        hbuf, W_c1, b_c1, W_c2, b_c2, out, N);
}